// VQC_29506425323622
// MI455X (gfx1250) — compile-verified
//
#include <hip/hip_runtime.h>
#include <math.h>

#define N_QUBITS 16
#define N_CLASS  10
#define DEPTH    6
#define BATCH    256
#define NSTATE   65536           // 2^16 amplitudes (real)
#define INV_SQRT2 0.70710678118654752440f

typedef __attribute__((ext_vector_type(2))) float v2f;
typedef __attribute__((ext_vector_type(8))) float v8f;

// Within-block CNOT permutation P = C_oddwithin ∘ C_even, as an index map:
// new[x] = old[ce(cow(x))].  Block-local bits: wire a+0 -> bit3 ... a+3 -> bit0.
// cow: ctrl bit2 flips bit1.  ce: ctrl bit3 flips bit2, ctrl bit1 flips bit0.
__device__ __forceinline__ int perm_ce_cow(int x) {
  int y = x ^ (((x >> 2) & 1) << 1);                       // cow
  y = y ^ ((((y >> 3) & 1) << 2) | (((y >> 1) & 1) << 0)); // ce
  return y;
}

__launch_bounds__(1024, 1)
__global__ void vqc_kernel(const float* __restrict__ X,
                           const float* __restrict__ W,
                           float* __restrict__ out) {
  __shared__ float st[NSTATE];              // 256 KB state vector
  __shared__ float mats[DEPTH * 4 * 256];   // 24 KB: 16x16 per (layer, block)
  __shared__ float T[64];                   // 4 block product tables
  __shared__ float amp[32];                 // amp[bit*16 + wire]
  __shared__ float red[32 * N_CLASS];       // wave partial sums

  const int tid  = threadIdx.x;
  const int b    = blockIdx.x;
  const int lane = tid & 31;
  const int wave = tid >> 5;
  const int hi   = (lane >> 4) & 1;
  const int row  = lane & 15;

  // ---- Build per-(layer, block) 16x16 matrices: M = P * (RY x RY x RY x RY)
  // (P folded for all but the last layer). Thread (kq, p) simulates column p.
  if (tid < DEPTH * 4 * 16) {
    const int kq = tid >> 4;          // 0..23
    const int k  = kq >> 2, q = kq & 3;
    const int p  = tid & 15;
    float v[16];
#pragma unroll
    for (int x = 0; x < 16; ++x) v[x] = (x == p) ? 1.0f : 0.0f;
#pragma unroll
    for (int t = 0; t < 4; ++t) {               // wire 4q+t -> local bit 3-t
      const float ang = 0.5f * W[k * 16 + 4 * q + t];
      const float c = cosf(ang), s = sinf(ang);
      const int bb = 3 - t;
#pragma unroll
      for (int x = 0; x < 16; ++x) {
        if (((x >> bb) & 1) == 0) {
          const int x1 = x | (1 << bb);
          const float a0 = v[x], a1 = v[x1];
          v[x]  = c * a0 - s * a1;
          v[x1] = s * a0 + c * a1;
        }
      }
    }
#pragma unroll
    for (int x = 0; x < 16; ++x) {
      const float val = (k < DEPTH - 1) ? v[perm_ce_cow(x)] : v[x];
      mats[(kq * 16 + x) * 16 + p] = val;
    }
  }

  // ---- Encoding: per-wire factors of RY(x/2) * H |0>  (product state)
  if (tid < 16) {
    const float ang = 0.5f * X[b * 16 + tid];
    const float c = cosf(ang), s = sinf(ang);
    amp[tid]      = (c - s) * INV_SQRT2;   // bit = 0
    amp[16 + tid] = (c + s) * INV_SQRT2;   // bit = 1
  }
  __syncthreads();

  // Block product tables with first layer's within-block perm folded in.
  if (tid < 64) {
    const int q = tid >> 4, j = tid & 15;
    const int jp = perm_ce_cow(j);
    float prod = 1.0f;
#pragma unroll
    for (int t = 0; t < 4; ++t) {
      const int bit = (jp >> (3 - t)) & 1;
      prod *= amp[bit * 16 + (4 * q + t)];
    }
    T[tid] = prod;
  }
  __syncthreads();

  // Fill the product state: psi[i] = prod over blocks of T_q[j_q(i)]
  for (int e = 0; e < 64; ++e) {
    const int i = (e << 10) | tid;
    st[i] = T[(i >> 12) & 15] * T[16 + ((i >> 8) & 15)] *
            T[32 + ((i >> 4) & 15)] * T[48 + (i & 15)];
  }
  __syncthreads();

  // ---- Layers: [cross-block CNOTs] then 4 per-block WMMA passes
  for (int k = 0; k < DEPTH; ++k) {
    // Cross CNOTs (3,4),(7,8),(11,12): wire w -> bit 15-w.
    // Involution: pair-swap, owner = smaller index, race-free.
    for (int e = 0; e < 64; ++e) {
      const int i = (e << 10) | tid;
      const int m = (((i >> 12) & 1) << 11) | (((i >> 8) & 1) << 7) |
                    (((i >> 4) & 1) << 3);
      const int p = i ^ m;
      if (p > i) { const float a = st[i]; st[i] = st[p]; st[p] = a; }
    }
    __syncthreads();

#pragma unroll
    for (int q = 0; q < 4; ++q) {
      const int sh = 12 - 4 * q;         // log2(inner size)
      const int I  = 1 << sh;
      // A fragments: A[row][4*kk + 2*hi + {0,1}] (16x4 f32 WMMA layout)
      const float* M = &mats[(k * 4 + q) * 256 + row * 16];
      const v2f a0 = { M[0  + 2 * hi], M[1  + 2 * hi] };
      const v2f a1 = { M[4  + 2 * hi], M[5  + 2 * hi] };
      const v2f a2 = { M[8  + 2 * hi], M[9  + 2 * hi] };
      const v2f a3 = { M[12 + 2 * hi], M[13 + 2 * hi] };
      const int j0 = 2 * hi;
      for (int t8 = 0; t8 < 8; ++t8) {
        const int tile  = wave * 8 + t8;       // 256 tiles of 16 columns
        const int c     = tile * 16 + row;     // this lane's column
        const int inner = c & (I - 1);
        const int base  = ((c & ~(I - 1)) << 4) | inner; // elem (outer, j=0, inner)
        v8f acc = {0.f, 0.f, 0.f, 0.f, 0.f, 0.f, 0.f, 0.f};
        {
          v2f bf = { st[base + (j0)      * I], st[base + (j0 + 1)  * I] };
          acc = __builtin_amdgcn_wmma_f32_16x16x4_f32(false, a0, false, bf, (short)0, acc, false, false);
        }
        {
          v2f bf = { st[base + (4 + j0)  * I], st[base + (5 + j0)  * I] };
          acc = __builtin_amdgcn_wmma_f32_16x16x4_f32(false, a1, false, bf, (short)0, acc, false, false);
        }
        {
          v2f bf = { st[base + (8 + j0)  * I], st[base + (9 + j0)  * I] };
          acc = __builtin_amdgcn_wmma_f32_16x16x4_f32(false, a2, false, bf, (short)0, acc, false, false);
        }
        {
          v2f bf = { st[base + (12 + j0) * I], st[base + (13 + j0) * I] };
          acc = __builtin_amdgcn_wmma_f32_16x16x4_f32(false, a3, false, bf, (short)0, acc, false, false);
        }
#pragma unroll
        for (int v = 0; v < 8; ++v)
          st[base + (v + 8 * hi) * I] = acc[v];   // D row = v + 8*hi, col = lane&15
      }
      __syncthreads();
    }
  }

  // ---- <Z> expectations on wires 0..9 (bit 15-pos)
  float acc[N_CLASS];
#pragma unroll
  for (int p = 0; p < N_CLASS; ++p) acc[p] = 0.0f;
  for (int e = 0; e < 64; ++e) {
    const int i = (e << 10) | tid;
    const float ps = st[i];
    const float p2 = ps * ps;
#pragma unroll
    for (int p = 0; p < N_CLASS; ++p)
      acc[p] += ((i >> (15 - p)) & 1) ? -p2 : p2;
  }
#pragma unroll
  for (int p = 0; p < N_CLASS; ++p)
    for (int m = 16; m >= 1; m >>= 1)
      acc[p] += __shfl_xor(acc[p], m, 32);
  if (lane == 0) {
#pragma unroll
    for (int p = 0; p < N_CLASS; ++p) red[wave * N_CLASS + p] = acc[p];
  }
  __syncthreads();
  if (tid < N_CLASS) {
    float s = 0.0f;
    for (int w = 0; w < 32; ++w) s += red[w * N_CLASS + tid];
    out[b * N_CLASS + tid] = s;
  }
}

extern "C" void kernel_launch(void* const* d_in, const int* in_sizes, int n_in,
                              void* d_out, int out_size, void* d_ws, size_t ws_size,
                              hipStream_t stream) {
  (void)in_sizes; (void)n_in; (void)d_ws; (void)ws_size; (void)out_size;
  const float* X = (const float*)d_in[0];   // (256, 16) f32
  const float* W = (const float*)d_in[1];   // (6, 16)  f32
  float* out = (float*)d_out;               // (256, 10) f32
  vqc_kernel<<<dim3(BATCH), dim3(1024), 0, stream>>>(X, W, out);
}